// HawkesLandmarkAttention_6674379178503
// MI455X (gfx1250) — compile-verified
//
#include <hip/hip_runtime.h>
#include <hip/hip_bf16.h>
#include <float.h>

typedef float v2f __attribute__((ext_vector_type(2)));
typedef float v8f __attribute__((ext_vector_type(8)));

#define B_ 16
#define L_ 8192
#define D_ 512
#define H_ 8
#define DH_ 64
#define K_ 100
#define IDX_STRIDE 128

__device__ inline v8f wmma_f32x4(v2f a, v2f b, v8f c) {
    // V_WMMA_F32_16X16X4_F32 : D = A(16x4 f32) * B(4x16 f32) + C(16x16 f32)
    return __builtin_amdgcn_wmma_f32_16x16x4_f32(
        /*neg_a=*/false, a, /*neg_b=*/false, b,
        /*c_mod=*/(short)0, c, /*reuse_a=*/false, /*reuse_b=*/false);
}

// ---------------------------------------------------------------------------
// 1) Top-k (k=100) per batch with JAX tie-breaking (lower index first on tie),
//    indices then sorted ascending (matches lax.top_k + jnp.sort).
// ---------------------------------------------------------------------------
__global__ void topk_kernel(const float* __restrict__ hw, int* __restrict__ idxOut) {
    __shared__ float w[L_];
    __shared__ float rv[256];
    __shared__ int   ri[256];
    __shared__ int   sel[IDX_STRIDE];

    const int b = blockIdx.x;
    const int t = threadIdx.x;
    const float* src = hw + (size_t)b * L_;

    for (int i = t; i < L_; i += 256) w[i] = src[i];
    __syncthreads();

    for (int it = 0; it < K_; ++it) {
        float bv = -FLT_MAX;
        int   bi = 0x7fffffff;
        for (int i = t; i < L_; i += 256) {
            float v = w[i];
            if (v > bv || (v == bv && i < bi)) { bv = v; bi = i; }
        }
        rv[t] = bv; ri[t] = bi;
        __syncthreads();
        for (int s = 128; s > 0; s >>= 1) {
            if (t < s) {
                float v2 = rv[t + s]; int i2 = ri[t + s];
                if (v2 > rv[t] || (v2 == rv[t] && i2 < ri[t])) { rv[t] = v2; ri[t] = i2; }
            }
            __syncthreads();
        }
        if (t == 0) { sel[it] = ri[0]; w[ri[0]] = -FLT_MAX; }
        __syncthreads();
    }

    if (t == 0) { // insertion sort, ascending indices
        for (int i = 1; i < K_; ++i) {
            int key = sel[i]; int j = i - 1;
            while (j >= 0 && sel[j] > key) { sel[j + 1] = sel[j]; --j; }
            sel[j + 1] = key;
        }
    }
    __syncthreads();
    if (t < K_) idxOut[b * IDX_STRIDE + t] = sel[t];
}

// ---------------------------------------------------------------------------
// 2) K & V projections: per wave, one 16x16 tile of K and V.
//    A rows gathered through landmark indices. Rows m >= 100 load from a
//    CLAMPED valid index (no divergent branch in the inner loop): since row m
//    of A only affects row m of C, and rows >= 100 are never written back,
//    the garbage is harmless. Inner loop is pure load + wmma.
//    C[m,n] = sum_d A[m,d] * W[n,d]  (+ bias[n])
// ---------------------------------------------------------------------------
__global__ void proj_kv_kernel(const float* __restrict__ skip,
                               const int*   __restrict__ idx,
                               const float* __restrict__ Wk, const float* __restrict__ bk,
                               const float* __restrict__ Wv, const float* __restrict__ bv,
                               float* __restrict__ Kout, float* __restrict__ Vout) {
    const int lane = threadIdx.x;          // 0..31 (wave32)
    const int n0   = blockIdx.x * 16;      // output column tile
    const int m0   = blockIdx.y * 16;      // landmark row tile (7 tiles cover 100)
    const int b    = blockIdx.z;
    const int row  = lane & 15;
    const int koff = (lane >> 4) * 2;

    const int gm  = m0 + row;                       // A row handled by this lane
    const int gmc = gm < (K_ - 1) ? gm : (K_ - 1);  // clamp padded rows to a valid one
    const int gi  = idx[b * IDX_STRIDE + gmc];
    const float* arow  = skip + ((size_t)b * L_ + (size_t)gi) * D_;
    const float* wkrow = Wk + (size_t)(n0 + row) * D_; // B col n = lane&15
    const float* wvrow = Wv + (size_t)(n0 + row) * D_;

    v8f ck = {}; v8f cv = {};
    for (int k0 = 0; k0 < D_; k0 += 4) {
        v2f a;   a.x   = arow[k0 + koff];  a.y   = arow[k0 + koff + 1];
        v2f bk2; bk2.x = wkrow[k0 + koff]; bk2.y = wkrow[k0 + koff + 1];
        v2f bv2; bv2.x = wvrow[k0 + koff]; bv2.y = wvrow[k0 + koff + 1];
        ck = wmma_f32x4(a, bk2, ck);
        cv = wmma_f32x4(a, bv2, cv);
    }

    const int nn    = n0 + row;
    const int mbase = m0 + (lane >> 4) * 8;
    const float bkb = bk[nn];
    const float bvb = bv[nn];
    #pragma unroll
    for (int r = 0; r < 8; ++r) {
        int mm = mbase + r;
        if (mm < K_) {
            Kout[((size_t)b * K_ + mm) * D_ + nn] = ck[r] + bkb;
            Vout[((size_t)b * K_ + mm) * D_ + nn] = cv[r] + bvb;
        }
    }
}

// ---------------------------------------------------------------------------
// 3) Generic 16-row projection: out(16x512) = A(16x512) @ W^T + bias.
//    Used for Q (A = last tokens across the 16 batches; batch == M) and for
//    the final output projection.
// ---------------------------------------------------------------------------
__global__ void proj16_kernel(const float* __restrict__ Abase, long long rowStride,
                              const float* __restrict__ W, const float* __restrict__ bias,
                              float* __restrict__ out) {
    const int lane = threadIdx.x;          // 0..31
    const int n0   = blockIdx.x * 16;
    const int row  = lane & 15;
    const int koff = (lane >> 4) * 2;

    const float* arow = Abase + (size_t)row * (size_t)rowStride;
    const float* wrow = W + (size_t)(n0 + row) * D_;

    v8f c = {};
    for (int k0 = 0; k0 < D_; k0 += 4) {
        v2f a;  a.x  = arow[k0 + koff]; a.y  = arow[k0 + koff + 1];
        v2f bb; bb.x = wrow[k0 + koff]; bb.y = wrow[k0 + koff + 1];
        c = wmma_f32x4(a, bb, c);
    }

    const int nn    = n0 + row;
    const int mbase = (lane >> 4) * 8;
    const float bv  = bias[nn];
    #pragma unroll
    for (int r = 0; r < 8; ++r) {
        int mm = mbase + r;     // 0..15 == batch row
        out[(size_t)mm * D_ + nn] = c[r] + bv;
    }
}

// ---------------------------------------------------------------------------
// 4) Softmax attention per (batch, head): scores(100) -> softmax -> attn(64).
// ---------------------------------------------------------------------------
__global__ void attn_kernel(const float* __restrict__ Qa,
                            const float* __restrict__ Ka,
                            const float* __restrict__ Va,
                            float* __restrict__ attn) {
    const int bh = blockIdx.x;
    const int b  = bh >> 3;
    const int h  = bh & 7;
    const int t  = threadIdx.x;            // 0..127

    __shared__ float q[DH_];
    __shared__ float sc[128];
    __shared__ float red[2];

    if (t < DH_) q[t] = Qa[(size_t)b * D_ + h * DH_ + t];
    __syncthreads();

    if (t < K_) {
        const float* krow = Ka + ((size_t)b * K_ + t) * D_ + h * DH_;
        float s = 0.0f;
        for (int d = 0; d < DH_; ++d) s += q[d] * krow[d];
        sc[t] = s * 0.125f;                // dh^-0.5 = 1/8
    }
    __syncthreads();

    if (t == 0) {
        float mx = sc[0];
        for (int i = 1; i < K_; ++i) mx = fmaxf(mx, sc[i]);
        red[0] = mx;
    }
    __syncthreads();

    if (t < K_) sc[t] = expf(sc[t] - red[0]);
    __syncthreads();

    if (t == 0) {
        float s = 0.0f;
        for (int i = 0; i < K_; ++i) s += sc[i];
        red[1] = 1.0f / s;
    }
    __syncthreads();

    if (t < DH_) {
        const float* vbase = Va + (size_t)b * K_ * D_ + h * DH_ + t;
        float acc = 0.0f;
        for (int k = 0; k < K_; ++k) acc += sc[k] * vbase[(size_t)k * D_];
        attn[(size_t)b * D_ + h * DH_ + t] = acc * red[1];
    }
}

// ---------------------------------------------------------------------------
// Host-side launch
// ---------------------------------------------------------------------------
extern "C" void kernel_launch(void* const* d_in, const int* in_sizes, int n_in,
                              void* d_out, int out_size, void* d_ws, size_t ws_size,
                              hipStream_t stream) {
    const float* mamba = (const float*)d_in[0];
    const float* skip  = (const float*)d_in[1];
    const float* hw    = (const float*)d_in[2];
    const float* Wq    = (const float*)d_in[3];
    const float* bq    = (const float*)d_in[4];
    const float* Wk    = (const float*)d_in[5];
    const float* bk    = (const float*)d_in[6];
    const float* Wv    = (const float*)d_in[7];
    const float* bv    = (const float*)d_in[8];
    const float* Wo    = (const float*)d_in[9];
    const float* bo    = (const float*)d_in[10];
    float* out = (float*)d_out;

    // Workspace layout (~6.7 MB)
    int*   idx  = (int*)d_ws;                                 // 16*128 ints
    float* Qa   = (float*)((char*)d_ws + B_ * IDX_STRIDE * sizeof(int));
    float* attn = Qa + (size_t)B_ * D_;                       // 16*512
    float* Ka   = attn + (size_t)B_ * D_;                     // 16*100*512
    float* Va   = Ka + (size_t)B_ * K_ * D_;                  // 16*100*512

    // 1) top-k landmark selection
    topk_kernel<<<B_, 256, 0, stream>>>(hw, idx);

    // 2) Q projection: the 16 batch queries form one 16-row WMMA M-tile
    proj16_kernel<<<D_ / 16, 32, 0, stream>>>(
        mamba + (size_t)(L_ - 1) * D_, (long long)L_ * D_, Wq, bq, Qa);

    // 3) K/V projections (gather landmarks through idx)
    dim3 gkv(D_ / 16, (K_ + 15) / 16, B_);
    proj_kv_kernel<<<gkv, 32, 0, stream>>>(skip, idx, Wk, bk, Wv, bv, Ka, Va);

    // 4) softmax attention
    attn_kernel<<<B_ * H_, 128, 0, stream>>>(Qa, Ka, Va, attn);

    // 5) output projection -> d_out (16x512)
    proj16_kernel<<<D_ / 16, 32, 0, stream>>>(attn, (long long)D_, Wo, bo, out);
}